// Segger_71219147702915
// MI455X (gfx1250) — compile-verified
//
#include <hip/hip_runtime.h>
#include <string.h>

// ---------------- problem constants (match reference) ----------------
#define NTX   50000
#define NBD   5000
#define ETT   800000
#define ETB   100000
#define HEADS 3
#define HC    96          // HEADS * 32

typedef float v2f __attribute__((ext_vector_type(2)));
typedef float v8f __attribute__((ext_vector_type(8)));
typedef unsigned int u32x4 __attribute__((ext_vector_type(4)));
typedef int i32x8 __attribute__((ext_vector_type(8)));
typedef int i32x4 __attribute__((ext_vector_type(4)));

__device__ __forceinline__ float lrelu(float x, float s) { return x > 0.f ? x : x * s; }

// monotone float <-> uint mapping for atomic max on floats
__device__ __forceinline__ unsigned f2ord(float f) {
    unsigned b = __float_as_uint(f);
    return (b & 0x80000000u) ? ~b : (b | 0x80000000u);
}
__device__ __forceinline__ float ord2f(unsigned u) {
    return (u & 0x80000000u) ? __uint_as_float(u & 0x7fffffffu) : __uint_as_float(~u);
}
static inline unsigned f2ord_host(float f) {
    unsigned b; memcpy(&b, &f, 4);
    return (b & 0x80000000u) ? ~b : (b | 0x80000000u);
}

// ---------------- TDM: DMA a 2D fp32 block W[rows][cols] into LDS ----------------
// Descriptor layout per CDNA5 ISA (2D, data_size=4B,
// tile == tensor, groups 2/3 unused). Issued by one wave; tracked via TENSORcnt.
// This toolchain's builtin takes 6 args (clang-23 / therock-10.0 form).
__device__ __forceinline__ void tdm_load_block(const float* W, unsigned ldsOff,
                                               int rows, int cols) {
    unsigned long long ga = (unsigned long long)(size_t)W;
    u32x4 g0;
    g0[0] = 1u;                                         // count=1 (valid user D#)
    g0[1] = ldsOff;                                     // lds_addr (bytes)
    g0[2] = (unsigned)(ga & 0xffffffffu);               // global_addr[31:0]
    g0[3] = (unsigned)((ga >> 32) & 0x01ffffffu)        // global_addr[56:32]
          | (2u << 30);                                 // type = 2 ("image")
    i32x8 g1;
    g1[0] = 2 << 16;                                    // data_size=2 (4B); mask/flags=0
    g1[1] = (cols & 0xffff) << 16;                      // tensor_dim0[15:0]
    g1[2] = (cols >> 16) | ((rows & 0xffff) << 16);     // tensor_dim0[31:16] | tensor_dim1[15:0]
    g1[3] = (rows >> 16) | (cols << 16);                // tensor_dim1[31:16] | tile_dim0
    g1[4] = rows;                                       // tile_dim1 | tile_dim2=0
    g1[5] = cols;                                       // tensor_dim0_stride[31:0]
    g1[6] = 0;                                          // stride hi | dim1_stride lo
    g1[7] = 0;
    i32x4 z4 = {0, 0, 0, 0};                            // groups 2/3: unused (2D tensor)
    i32x8 z8 = {0, 0, 0, 0, 0, 0, 0, 0};
    __builtin_amdgcn_tensor_load_to_lds(g0, g1, z4, z4, z8, 0);
}

// ---------------- utility kernels ----------------
__global__ void k_fill_u32(unsigned* p, unsigned v, int n) {
    int i = blockIdx.x * blockDim.x + threadIdx.x;
    if (i < n) p[i] = v;
}

__global__ void k_embed(const int* __restrict__ tok, const float* __restrict__ emb,
                        float* __restrict__ h, int n) {
    int i = blockIdx.x * blockDim.x + threadIdx.x;
    if (i < n * 16) {
        int r = i >> 4, j = i & 15;
        h[i] = lrelu(emb[tok[r] * 16 + j], 0.01f);
    }
}

// ---------------- fp32 WMMA GEMM: Y[N,NO] = act(X[N,K] @ W[K,NO] + bias) ----------------
// Stage 1: Tensor Data Mover DMAs W row-major into LDS (wave 0 issues; TENSORcnt).
// Stage 2: one-shot LDS->LDS transpose into a padded tile (stride K+4: conflict-free
//          column reads since (K+4)*delta % 64 == 0 only when delta % 16 == 0).
// Main loop: one wave computes a 16 x NO row strip; per 4-wide K step one global b64
// A-fragment is reused across NT column tiles, B-fragments are contiguous ds_load_b64.
// V_WMMA_F32_16X16X4_F32 lane layout:
//   A (16x4): m=lane&15, kb=(lane>>4)*2 -> a = {A[m][kb], A[m][kb+1]}
//   B (4x16): n=lane&15, kb=(lane>>4)*2 -> b = {B[kb][n],  B[kb+1][n]}
//   C/D     : reg r -> row (r + 8*(lane>>4)), col = lane&15
template <int K, int NT>
__global__ void k_gemm_wmma(const float* __restrict__ X, int N,
                            const float* __restrict__ W, const float* __restrict__ bias,
                            float* __restrict__ Y, float slope /* <0: no act */) {
    constexpr int NO  = NT * 16;
    constexpr int LDK = K + 4;
    __shared__ float wraw[K * NO];             // TDM destination, row-major [K][NO]
    __shared__ float wt[NO * LDK];             // transposed [NO][K+4]

    if (threadIdx.x < 32) {                    // wave 0 drives the TDM
        tdm_load_block(W, (unsigned)(size_t)wraw, K, NO);
        __builtin_amdgcn_s_wait_tensorcnt(0);
    }
    __syncthreads();
    for (int idx = threadIdx.x; idx < K * NO; idx += blockDim.x) {
        int k = idx / NO, n = idx - k * NO;
        wt[n * LDK + k] = wraw[idx];           // LDS->LDS transpose
    }
    __syncthreads();

    const int lane  = threadIdx.x & 31;
    const int wave  = threadIdx.x >> 5;
    const int tileM = (blockIdx.x * (blockDim.x >> 5) + wave) * 16;
    if (tileM >= N) return;                    // uniform per-wave

    const int m  = lane & 15;
    const int kb = (lane >> 4) << 1;
    int row = tileM + m;
    if (row > N - 1) row = N - 1;              // clamp: unconditional loads, stores guarded
    const float* xp = X + (size_t)row * K;

    v8f c[NT];
    v8f zc = {};
#pragma unroll
    for (int t = 0; t < NT; ++t) c[t] = zc;

#pragma unroll
    for (int k = 0; k < K; k += 4) {
        v2f a = *(const v2f*)(xp + k + kb);    // global_load_b64, immediate offset
#pragma unroll
        for (int t = 0; t < NT; ++t) {
            v2f b = *(const v2f*)(wt + (t * 16 + m) * LDK + k + kb);  // ds_load_b64
            c[t] = __builtin_amdgcn_wmma_f32_16x16x4_f32(false, a, false, b, (short)0,
                                                         c[t], false, false);
        }
    }

    const int rbase = tileM + ((lane >> 4) << 3);
#pragma unroll
    for (int t = 0; t < NT; ++t) {
        float bv = bias ? bias[t * 16 + m] : 0.f;
#pragma unroll
        for (int r = 0; r < 8; ++r) {
            int orow = rbase + r;
            if (orow < N) {
                float v = c[t][r] + bv;
                if (slope >= 0.f) v = lrelu(v, slope);
                Y[(size_t)orow * NO + t * 16 + m] = v;
            }
        }
    }
}

// ---------------- GATv2 edge passes ----------------
// pass 1: logits[e,h] = att[h] . leaky_relu(xl[src]+xr[dst], 0.2); segment max into m
__global__ void k_logits_max(const float* __restrict__ xl, const float* __restrict__ xr,
                             const int* __restrict__ src, const int* __restrict__ dst,
                             int nplain, int ntot, const float* __restrict__ att,
                             float* __restrict__ exbuf, unsigned* __restrict__ m) {
    int idx = blockIdx.x * blockDim.x + threadIdx.x;
    if (idx >= ntot * HEADS) return;
    int e = idx / HEADS, h = idx - e * HEADS;
    int s, d;
    if (e < nplain) { s = src[e]; d = dst[e]; } else { s = e - nplain; d = s; }  // self loop
    if ((idx & 255) == 0 && e + 2048 < nplain) {       // stream-ahead hint (global_prefetch_b8)
        __builtin_prefetch(src + e + 2048, 0, 1);
        __builtin_prefetch(dst + e + 2048, 0, 1);
    }
    const float4* pa = (const float4*)(xl + (size_t)s * HC + h * 32);
    const float4* pb = (const float4*)(xr + (size_t)d * HC + h * 32);
    const float4* pt = (const float4*)(att + h * 32);
    float l = 0.f;
#pragma unroll
    for (int q = 0; q < 8; ++q) {
        float4 A = pa[q], B = pb[q], T = pt[q];
        l += T.x * lrelu(A.x + B.x, 0.2f) + T.y * lrelu(A.y + B.y, 0.2f)
           + T.z * lrelu(A.z + B.z, 0.2f) + T.w * lrelu(A.w + B.w, 0.2f);
    }
    exbuf[idx] = l;
    atomicMax(&m[d * HEADS + h], f2ord(l));
}

// pass 2: ex = exp(logit - m[dst]); segment-sum into denom
__global__ void k_exp_denom(const int* __restrict__ dst, int nplain, int ntot,
                            float* __restrict__ exbuf, const unsigned* __restrict__ m,
                            float* __restrict__ den) {
    int idx = blockIdx.x * blockDim.x + threadIdx.x;
    if (idx >= ntot * HEADS) return;
    int e = idx / HEADS, h = idx - e * HEADS;
    int d = (e < nplain) ? dst[e] : (e - nplain);
    float v = __expf(exbuf[idx] - ord2f(m[d * HEADS + h]));
    exbuf[idx] = v;
    atomicAdd(&den[d * HEADS + h], v);
}

// pass 3: agg[dst] += (ex/denom[dst]) * xl[src]  (one thread per edge x float4 chunk)
__global__ void k_scatter(const int* __restrict__ src, const int* __restrict__ dst,
                          int nplain, int ntot, const float* __restrict__ exbuf,
                          const float* __restrict__ den, const float* __restrict__ xl,
                          float* __restrict__ agg) {
    int idx = blockIdx.x * blockDim.x + threadIdx.x;
    if (idx >= ntot * 24) return;
    int e = idx / 24, q = idx - e * 24;
    int j = q * 4, h = j >> 5;
    int s, d;
    if (e < nplain) { s = src[e]; d = dst[e]; } else { s = e - nplain; d = s; }
    float a  = exbuf[e * HEADS + h] / (den[d * HEADS + h] + 1e-16f);
    float4 X = *(const float4*)(xl + (size_t)s * HC + j);
    float* o = agg + (size_t)d * HC + j;
    atomicAdd(o + 0, a * X.x); atomicAdd(o + 1, a * X.y);
    atomicAdd(o + 2, a * X.z); atomicAdd(o + 3, a * X.w);
}

// h = leaky_relu(agg + gat_bias + skip, 0.01)
__global__ void k_combine(float* __restrict__ h, const float* __restrict__ agg,
                          const float* __restrict__ gbias, const float* __restrict__ skip,
                          int rows) {
    int idx = blockIdx.x * blockDim.x + threadIdx.x;
    if (idx >= rows * HC) return;
    int j = idx % HC;
    h[idx] = lrelu(agg[idx] + gbias[j] + skip[idx], 0.01f);
}

// ---------------- host launcher ----------------
extern "C" void kernel_launch(void* const* d_in, const int* in_sizes, int n_in,
                              void* d_out, int out_size, void* d_ws, size_t ws_size,
                              hipStream_t stream) {
    (void)in_sizes; (void)n_in; (void)out_size; (void)ws_size;

    // positional inputs (setup_inputs dict order)
    const int*   x_tx   = (const int*)d_in[0];
    const float* x_bd   = (const float*)d_in[1];
    const int*   tt_src = (const int*)d_in[2];           // edge_tt row 0
    const int*   tt_dst = tt_src + ETT;                  // edge_tt row 1
    const int*   tb_src = (const int*)d_in[3];
    const int*   tb_dst = (const int*)d_in[4];

    // params flattened via jax tree_leaves (sorted dict keys):
    // bd_init{W,b}, emb, final_bd{W,b}, final_tx{W,b},
    // layers[i]{ lin_bd{W,b}, lin_tx{W,b}, tb{Wl,Wr,att,bias}, tt{Wl,Wr,att,bias} }
    const float* bdW = (const float*)d_in[5];
    const float* bdB = (const float*)d_in[6];
    const float* emb = (const float*)d_in[7];
    const float* fbW = (const float*)d_in[8];
    const float* fbB = (const float*)d_in[9];
    const float* ftW = (const float*)d_in[10];
    const float* ftB = (const float*)d_in[11];

    // workspace carve (floats), all chunks 16B aligned
    float* ws = (float*)d_ws;
    size_t o = 0;
    float* h_tx    = ws + o; o += (size_t)NTX * HC;
    float* h_bd    = ws + o; o += (size_t)NBD * HC;
    float* xl_tt   = ws + o; o += (size_t)NTX * HC;
    float* xr_tt   = ws + o; o += (size_t)NTX * HC;
    float* xl_tb   = ws + o; o += (size_t)NTX * HC;
    float* xr_tb   = ws + o; o += (size_t)NBD * HC;
    float* skip_tx = ws + o; o += (size_t)NTX * HC;
    float* skip_bd = ws + o; o += (size_t)NBD * HC;
    float* agg_tx  = ws + o; o += (size_t)NTX * HC;
    float* agg_bd  = ws + o; o += (size_t)NBD * HC;
    unsigned* m_tt = (unsigned*)(ws + o); o += (size_t)NTX * HEADS;
    float* den_tt  = ws + o; o += (size_t)NTX * HEADS;
    float* ex_tt   = ws + o; o += (size_t)(ETT + NTX) * HEADS;
    unsigned* m_tb = (unsigned*)(ws + o); o += (size_t)NBD * HEADS;
    float* den_tb  = ws + o; o += (size_t)NBD * HEADS;
    float* ex_tb   = ws + o; o += (size_t)ETB * HEADS;

    const unsigned NEG_ORD = f2ord_host(-1e30f);
    auto cdiv = [](long long a, long long b) { return (unsigned)((a + b - 1) / b); };
    // 4 waves / block, each wave computes a 16 x NO row strip
    auto gemm = [&](const float* X, int N, int K, const float* W, const float* B,
                    float* Y, int NO, float slope) {
        dim3 g(cdiv(N, 64));
        if (K == 16 && NO == 96)
            k_gemm_wmma<16, 6><<<g, 128, 0, stream>>>(X, N, W, B, Y, slope);
        else if (K == 96 && NO == 96)
            k_gemm_wmma<96, 6><<<g, 128, 0, stream>>>(X, N, W, B, Y, slope);
        else if (K == 32 && NO == 16)
            k_gemm_wmma<32, 1><<<g, 128, 0, stream>>>(X, N, W, B, Y, slope);
        else if (K == 96 && NO == 32)
            k_gemm_wmma<96, 2><<<g, 128, 0, stream>>>(X, N, W, B, Y, slope);
    };
    auto fill = [&](unsigned* p, unsigned v, long long n) {
        k_fill_u32<<<cdiv(n, 256), 256, 0, stream>>>(p, v, (int)n);
    };

    // ---- init node features ----
    k_embed<<<cdiv((long long)NTX * 16, 256), 256, 0, stream>>>(x_tx, emb, h_tx, NTX);
    gemm(x_bd, NBD, 32, bdW, bdB, h_bd, 16, 0.01f);   // h_bd = lrelu(x_bd@W+b)

    const int NTT = ETT + NTX;  // tt edges incl. self loops
    for (int L = 0; L < 5; ++L) {
        const int K = (L == 0) ? 16 : 96;
        const int base = 12 + 12 * L;
        const float* linbdW = (const float*)d_in[base + 0];
        const float* linbdB = (const float*)d_in[base + 1];
        const float* lintxW = (const float*)d_in[base + 2];
        const float* lintxB = (const float*)d_in[base + 3];
        const float* tbWl   = (const float*)d_in[base + 4];
        const float* tbWr   = (const float*)d_in[base + 5];
        const float* tbAtt  = (const float*)d_in[base + 6];
        const float* tbBias = (const float*)d_in[base + 7];
        const float* ttWl   = (const float*)d_in[base + 8];
        const float* ttWr   = (const float*)d_in[base + 9];
        const float* ttAtt  = (const float*)d_in[base + 10];
        const float* ttBias = (const float*)d_in[base + 11];

        // dense transforms (WMMA fp32, weights via TDM)
        gemm(h_tx, NTX, K, ttWl,   nullptr, xl_tt,   HC, -1.f);
        gemm(h_tx, NTX, K, ttWr,   nullptr, xr_tt,   HC, -1.f);
        gemm(h_tx, NTX, K, tbWl,   nullptr, xl_tb,   HC, -1.f);
        gemm(h_bd, NBD, K, tbWr,   nullptr, xr_tb,   HC, -1.f);
        gemm(h_tx, NTX, K, lintxW, lintxB,  skip_tx, HC, -1.f);
        gemm(h_bd, NBD, K, linbdW, linbdB,  skip_bd, HC, -1.f);

        // reset segment buffers
        fill(m_tt, NEG_ORD, (long long)NTX * HEADS);
        fill((unsigned*)den_tt, 0u, (long long)NTX * HEADS);
        fill((unsigned*)agg_tx, 0u, (long long)NTX * HC);
        fill(m_tb, NEG_ORD, (long long)NBD * HEADS);
        fill((unsigned*)den_tb, 0u, (long long)NBD * HEADS);
        fill((unsigned*)agg_bd, 0u, (long long)NBD * HC);

        // tt attention (with self loops)
        k_logits_max<<<cdiv((long long)NTT * HEADS, 256), 256, 0, stream>>>(
            xl_tt, xr_tt, tt_src, tt_dst, ETT, NTT, ttAtt, ex_tt, m_tt);
        k_exp_denom<<<cdiv((long long)NTT * HEADS, 256), 256, 0, stream>>>(
            tt_dst, ETT, NTT, ex_tt, m_tt, den_tt);
        k_scatter<<<cdiv((long long)NTT * 24, 256), 256, 0, stream>>>(
            tt_src, tt_dst, ETT, NTT, ex_tt, den_tt, xl_tt, agg_tx);

        // tb attention (no self loops)
        k_logits_max<<<cdiv((long long)ETB * HEADS, 256), 256, 0, stream>>>(
            xl_tb, xr_tb, tb_src, tb_dst, ETB, ETB, tbAtt, ex_tb, m_tb);
        k_exp_denom<<<cdiv((long long)ETB * HEADS, 256), 256, 0, stream>>>(
            tb_dst, ETB, ETB, ex_tb, m_tb, den_tb);
        k_scatter<<<cdiv((long long)ETB * 24, 256), 256, 0, stream>>>(
            tb_src, tb_dst, ETB, ETB, ex_tb, den_tb, xl_tb, agg_bd);

        // residual combine + activation (overwrites h in place; old h no longer needed)
        k_combine<<<cdiv((long long)NTX * HC, 256), 256, 0, stream>>>(
            h_tx, agg_tx, ttBias, skip_tx, NTX);
        k_combine<<<cdiv((long long)NBD * HC, 256), 256, 0, stream>>>(
            h_bd, agg_bd, tbBias, skip_bd, NBD);
    }

    // final projections straight into d_out: [out_tx | out_bd]
    float* out = (float*)d_out;
    gemm(h_tx, NTX, 96, ftW, ftB, out, 32, -1.f);
    gemm(h_bd, NBD, 96, fbW, fbB, out + (size_t)NTX * 32, 32, -1.f);
}